// Reconstruct_74217034874913
// MI455X (gfx1250) — compile-verified
//
#include <hip/hip_runtime.h>
#include <stdint.h>

typedef unsigned int u32x4 __attribute__((ext_vector_type(4)));
typedef int          i32x8 __attribute__((ext_vector_type(8)));
typedef int          i32x4 __attribute__((ext_vector_type(4)));

#define BLOCK          256
#define PTS_PER_BLOCK  1024   // 2 float2 tiles of 8 KB each in LDS

__global__ __launch_bounds__(BLOCK) void Reconstruct_tdm_kernel(
    const float* __restrict__ p1, const float* __restrict__ p2,
    const float* __restrict__ Rm, const float* __restrict__ tv,
    const float* __restrict__ Ri, const float* __restrict__ Ri1,
    const float* __restrict__ Ri2, const float* __restrict__ T1p,
    const float* __restrict__ T2p, const float* __restrict__ Dp,
    float* __restrict__ out, int n)
{
    __shared__ float lds1[2 * PTS_PER_BLOCK];
    __shared__ float lds2[2 * PTS_PER_BLOCK];

    const int base = (int)blockIdx.x * PTS_PER_BLOCK;
    int cnt = n - base;
    if (cnt > PTS_PER_BLOCK) cnt = PTS_PER_BLOCK;
    const unsigned ne = 2u * (unsigned)cnt;   // elements (floats) in each tile

    // --- TDM: one wave issues two async tensor loads (Global -> LDS) ---
    if (threadIdx.x < 32) {
        const uint64_t ga1 = (uint64_t)(uintptr_t)(p1 + 2 * base);
        const uint64_t ga2 = (uint64_t)(uintptr_t)(p2 + 2 * base);
        // low 32 bits of a generic LDS pointer == workgroup-relative LDS byte address
        const unsigned l1 = (unsigned)(uintptr_t)&lds1[0];
        const unsigned l2 = (unsigned)(uintptr_t)&lds2[0];

        // D# group1: data_size=4B; tensor_dim0=ne; tensor_dim1=1; tile_dim0=ne;
        // tile_dim1=1; tile_dim2=0; tensor_dim0_stride=ne; no pad/iterate/multicast.
        i32x8 g1;
        g1[0] = (int)0x00020000u;                              // mask=0 | data_size=2 (4B)
        g1[1] = (int)((ne & 0xFFFFu) << 16);                   // tensor_dim0[15:0]
        g1[2] = (int)(((ne >> 16) & 0xFFFFu) | (1u << 16));    // tensor_dim0[31:16] | tensor_dim1[15:0]=1
        g1[3] = (int)((ne & 0xFFFFu) << 16);                   // tensor_dim1[31:16]=0 | tile_dim0=ne
        g1[4] = 1;                                             // tile_dim1=1, tile_dim2=0
        g1[5] = (int)ne;                                       // tensor_dim0_stride[31:0]
        g1[6] = 0;                                             // stride hi | tensor_dim1_stride lo
        g1[7] = 0;
        const i32x4 gz = {0, 0, 0, 0};                         // groups 2/3 unused (<=2D)
        const i32x8 gz8 = {0, 0, 0, 0, 0, 0, 0, 0};           // 6-arg form extra group

        u32x4 g0;
        g0[0] = 1u;                                            // count=1 valid descriptor
        g0[1] = l1;                                            // lds_addr
        g0[2] = (unsigned)ga1;                                 // global_addr[31:0]
        g0[3] = ((unsigned)(ga1 >> 32) & 0x01FFFFFFu) | 0x80000000u; // addr[56:32] | type=2
        __builtin_amdgcn_tensor_load_to_lds(g0, g1, gz, gz, gz8, 0);

        g0[1] = l2;
        g0[2] = (unsigned)ga2;
        g0[3] = ((unsigned)(ga2 >> 32) & 0x01FFFFFFu) | 0x80000000u;
        __builtin_amdgcn_tensor_load_to_lds(g0, g1, gz, gz, gz8, 0);

        __builtin_amdgcn_s_wait_tensorcnt(0);                  // TENSORcnt==0 -> LDS valid
    }
    __syncthreads();
    asm volatile("" ::: "memory");   // LDS written by TDM, not by visible stores

    // --- uniform constants -> SGPRs ---
    const float D0 = Dp[0], D1 = Dp[1];
    const float R00 = Rm[0], R01 = Rm[1], R02 = Rm[2];
    const float R10 = Rm[3], R11 = Rm[4], R12 = Rm[5];
    const float R20 = Rm[6], R21 = Rm[7], R22 = Rm[8];
    const float t0 = tv[0], t1 = tv[1], t2 = tv[2];
    const float A00 = Ri[0],  A01 = Ri[1],  A02 = Ri[2];
    const float A10 = Ri[3],  A11 = Ri[4],  A12 = Ri[5];
    const float A20 = Ri[6],  A21 = Ri[7],  A22 = Ri[8];
    const float B00 = Ri1[0], B01 = Ri1[1], B02 = Ri1[2];
    const float B10 = Ri1[3], B11 = Ri1[4], B12 = Ri1[5];
    const float B20 = Ri1[6], B21 = Ri1[7], B22 = Ri1[8];
    const float E00 = Ri2[0], E01 = Ri2[1], E02 = Ri2[2];
    const float E10 = Ri2[3], E11 = Ri2[4], E12 = Ri2[5];
    const float E20 = Ri2[6], E21 = Ri2[7], E22 = Ri2[8];
    const float T1x = T1p[0], T1y = T1p[1], T1z = T1p[2];
    const float T2x = T2p[0], T2y = T2p[1], T2z = T2p[2];

    const float2* q1p = (const float2*)lds1;
    const float2* q2p = (const float2*)lds2;
    float2* o1 = (float2*)out;                       // pred1: [0, 2N)
    float2* o2 = (float2*)(out + 2 * (size_t)n);     // pred2: [2N, 4N)
    float*  ot = out + 4 * (size_t)n;                // temp : [4N, 7N)

    #pragma unroll
    for (int k = 0; k < PTS_PER_BLOCK / BLOCK; ++k) {
        const int j = (int)threadIdx.x + k * BLOCK;
        if (j < cnt) {
            const float2 q1 = q1p[j];   // ds_load_b64, conflict-free
            const float2 q2 = q2p[j];

            const float ksi1 = q1.x / D0, eta1 = q1.y / D0;
            const float ksi2 = q2.x / D1, eta2 = q2.y / D1;

            // a = R[0] - ksi2*R[2];  b = R[1] - eta2*R[2]
            const float a0 = R00 - ksi2 * R20, a1 = R01 - ksi2 * R21, a2 = R02 - ksi2 * R22;
            const float b0 = R10 - eta2 * R20, b1 = R11 - eta2 * R21, b2 = R12 - eta2 * R22;
            const float at = a0 * t0 + a1 * t1 + a2 * t2;
            const float bt = b0 * t0 + b1 * t1 + b2 * t2;

            // ATA (symmetric 3x3) and ATB
            const float m00 = 1.0f + a0 * a0 + b0 * b0;
            const float m01 = a0 * a1 + b0 * b1;
            const float m02 = -ksi1 + a0 * a2 + b0 * b2;
            const float m11 = 1.0f + a1 * a1 + b1 * b1;
            const float m12 = -eta1 + a1 * a2 + b1 * b2;
            const float m22 = ksi1 * ksi1 + eta1 * eta1 + a2 * a2 + b2 * b2;
            const float v0 = a0 * at + b0 * bt;
            const float v1 = a1 * at + b1 * bt;
            const float v2 = a2 * at + b2 * bt;

            // 3x3 symmetric solve via adjugate
            const float c00 = m11 * m22 - m12 * m12;
            const float c01 = m02 * m12 - m01 * m22;
            const float c02 = m01 * m12 - m02 * m11;
            const float det = m00 * c00 + m01 * c01 + m02 * c02;
            const float c11 = m00 * m22 - m02 * m02;
            const float c12 = m01 * m02 - m00 * m12;
            const float c22 = m00 * m11 - m01 * m01;
            const float invdet = 1.0f / det;
            const float X = (c00 * v0 + c01 * v1 + c02 * v2) * invdet;
            const float Y = (c01 * v0 + c11 * v1 + c12 * v2) * invdet;
            const float Z = (c02 * v0 + c12 * v1 + c22 * v2) * invdet;

            // temp = R_inv @ (C - T1)
            const float d0 = X - T1x, d1 = Y - T1y, d2 = Z - T1z;
            const float w0 = A00 * d0 + A01 * d1 + A02 * d2;
            const float w1 = A10 * d0 + A11 * d1 + A12 * d2;
            const float w2 = A20 * d0 + A21 * d1 + A22 * d2;

            // ti1 = R_inv1 @ temp + T1 ; ti2 = R_inv2 @ temp + T2
            const float u1x = B00 * w0 + B01 * w1 + B02 * w2 + T1x;
            const float u1y = B10 * w0 + B11 * w1 + B12 * w2 + T1y;
            const float u1z = B20 * w0 + B21 * w1 + B22 * w2 + T1z;
            const float u2x = E00 * w0 + E01 * w1 + E02 * w2 + T2x;
            const float u2y = E10 * w0 + E11 * w1 + E12 * w2 + T2y;
            const float u2z = E20 * w0 + E21 * w1 + E22 * w2 + T2z;

            const int i = base + j;
            o1[i] = make_float2(u1x / u1z * D0, u1y / u1z * D0);
            o2[i] = make_float2(u2x / u2z * D1, u2y / u2z * D1);
            const size_t ti = 3 * (size_t)i;
            ot[ti + 0] = w0;
            ot[ti + 1] = w1;
            ot[ti + 2] = w2;
        }
    }
}

extern "C" void kernel_launch(void* const* d_in, const int* in_sizes, int n_in,
                              void* d_out, int out_size, void* d_ws, size_t ws_size,
                              hipStream_t stream) {
    (void)n_in; (void)out_size; (void)d_ws; (void)ws_size;
    const float* p1  = (const float*)d_in[0];
    const float* p2  = (const float*)d_in[1];
    const float* Rm  = (const float*)d_in[2];
    const float* tv  = (const float*)d_in[3];
    const float* Ri  = (const float*)d_in[4];
    const float* Ri1 = (const float*)d_in[5];
    const float* Ri2 = (const float*)d_in[6];
    const float* T1p = (const float*)d_in[7];
    const float* T2p = (const float*)d_in[8];
    const float* Dp  = (const float*)d_in[9];

    const int n = in_sizes[0] / 2;
    const int blocks = (n + PTS_PER_BLOCK - 1) / PTS_PER_BLOCK;
    Reconstruct_tdm_kernel<<<blocks, BLOCK, 0, stream>>>(
        p1, p2, Rm, tv, Ri, Ri1, Ri2, T1p, T2p, Dp, (float*)d_out, n);
}